// TemporalGraphConv_23184233464487
// MI455X (gfx1250) — compile-verified
//
#include <hip/hip_runtime.h>

typedef float v2f __attribute__((ext_vector_type(2)));
typedef float v8f __attribute__((ext_vector_type(8)));
typedef int   v4i __attribute__((ext_vector_type(4)));

#define WAVES_PER_BLOCK 4
#define KTILE 32                 // K elements staged per group (one 128B line/row)
#define ROWSTRIDE 36             // dwords; 144B: 16B-aligned, bank-conflict-free
#define STAGE_DW (16 * ROWSTRIDE)

// ---- CDNA5 async global->LDS copy (B128) + asynccnt wait ------------------
// Probe-derived signature: (AS1 v4i* src, AS3 v4i* dst, imm offset, imm cpol)
#if __has_builtin(__builtin_amdgcn_global_load_async_to_lds_b128)
#define ASYNC_CP_B128(gsrc, ldst)                                              \
  __builtin_amdgcn_global_load_async_to_lds_b128(                              \
      (__attribute__((address_space(1))) v4i*)(gsrc),                          \
      (__attribute__((address_space(3))) v4i*)(ldst), 0, 0)
#else
#define ASYNC_CP_B128(gsrc, ldst)                                              \
  asm volatile("global_load_async_to_lds_b128 %0, %1, off" ::                  \
                   "v"((unsigned)(uintptr_t)(                                  \
                       __attribute__((address_space(3))) void*)(ldst)),        \
               "v"(gsrc)                                                       \
               : "memory")
#endif

#if __has_builtin(__builtin_amdgcn_s_wait_asynccnt)
#define WAIT_ASYNC(n) __builtin_amdgcn_s_wait_asynccnt(n)
#else
#define WAIT_ASYNC(n) asm volatile("s_wait_asynccnt %0" ::"i"(n) : "memory")
#endif

// ---------------------------------------------------------------------------
// proj16: out[r, 0..15] = sum_k in[r,k] * W[k, 0..15]
//   One wave computes a 16-row tile via V_WMMA_F32_16X16X4_F32, K stepped by 4.
// ---------------------------------------------------------------------------
__global__ void proj16(const float* __restrict__ in, const float* __restrict__ W,
                       float* __restrict__ out, int R, int K) {
  const int lane = threadIdx.x & 31;
  const int wave = threadIdx.x >> 5;
  const int tile = blockIdx.x * WAVES_PER_BLOCK + wave;
  const int nTiles = R >> 4;
  if (tile >= nTiles) return;            // wave-uniform: EXEC stays all-ones

  const int rowBase = tile << 4;
  const int n    = lane & 15;            // M for A-frag, N for B/C-frag
  const int hi   = lane >> 4;            // 0: lanes 0-15, 1: lanes 16-31
  const int koff = hi << 1;              // upper-half lanes hold K+2,K+3

  const float* aRow = in + (size_t)(rowBase + n) * K + koff;
  const float* bCol = W + n;

  v8f c = {};
  for (int k = 0; k < K; k += 4) {
    v2f A = *(const v2f*)(aRow + k);
    v2f Bf;
    Bf.x = bCol[(k + koff) * 16];
    Bf.y = bCol[(k + koff + 1) * 16];
    c = __builtin_amdgcn_wmma_f32_16x16x4_f32(false, A, false, Bf,
                                              (short)0, c, false, false);
  }

  float* orow = out + ((size_t)rowBase + (hi << 3)) * 16 + n;
#pragma unroll
  for (int i = 0; i < 8; ++i) orow[i * 16] = c[i];
}

// ---------------------------------------------------------------------------
// agg16: out[b, r, 0..15] = relu( sum_m a[b,r,m] * h[b,m,0..15] + bias )
//   HBM-bound a-stream staged via double-buffered async global->LDS B128
//   copies (coalesced 128B/row), fragments read back as conflict-free
//   ds_load_b64 in the WMMA f32 A layout.
// ---------------------------------------------------------------------------
__global__ void agg16(const float* __restrict__ a, const float* __restrict__ h,
                      const float* __restrict__ bias, float* __restrict__ out,
                      int N, int totalTiles) {
  __shared__ float lds[WAVES_PER_BLOCK * 2 * STAGE_DW];

  const int lane = threadIdx.x & 31;
  const int wave = threadIdx.x >> 5;
  const int tile = blockIdx.x * WAVES_PER_BLOCK + wave;
  if (tile >= totalTiles) return;        // wave-uniform

  const int tilesPerBatch = N >> 4;
  const int b       = tile / tilesPerBatch;
  const int rowBase = (tile % tilesPerBatch) << 4;

  const int n    = lane & 15;
  const int hi   = lane >> 4;
  const int koff = hi << 1;

  const float* aTile = a + ((size_t)b * N + rowBase) * N;  // + row*N + k
  const float* hB    = h + (size_t)b * N * 16 + n;
  float* myLds = lds + wave * 2 * STAGE_DW;

  // chunk c = i*32 + lane: row = c>>3, kc = c&7  (8 lanes cover one 128B line)
  const int srow = lane >> 3;            // rows 0..3 base for i-steps of 4
  const int skc  = lane & 7;

  v8f c = {};

  // ---- prologue: stage group 0 into buffer 0
  {
    const float* g = aTile;
    float* dst = myLds;
#pragma unroll
    for (int i = 0; i < 4; ++i) {
      const int row = srow + i * 4;
      ASYNC_CP_B128(g + (size_t)row * N + skc * 4,
                    dst + row * ROWSTRIDE + skc * 4);
    }
  }

  int cur = 0;
  for (int m = 0; m < N; m += KTILE) {
    if (m + KTILE < N) {
      // stage next group into the other buffer, then wait for current one
      const float* g = aTile + (m + KTILE);
      float* dst = myLds + (cur ^ 1) * STAGE_DW;
#pragma unroll
      for (int i = 0; i < 4; ++i) {
        const int row = srow + i * 4;
        ASYNC_CP_B128(g + (size_t)row * N + skc * 4,
                      dst + row * ROWSTRIDE + skc * 4);
      }
      WAIT_ASYNC(4);                     // 4 outstanding = next group only
    } else {
      WAIT_ASYNC(0);                     // drain final group
    }
    asm volatile("" ::: "memory");       // keep ds reads below the wait

    const float* frag = myLds + cur * STAGE_DW + n * ROWSTRIDE + koff;
#pragma unroll
    for (int j = 0; j < 8; ++j) {
      v2f A = *(const v2f*)(frag + 4 * j);        // ds_load_b64, conflict-free
      v2f Bf;
      Bf.x = hB[(m + 4 * j + koff) * 16];         // h tiles: L2-resident
      Bf.y = hB[(m + 4 * j + koff + 1) * 16];
      c = __builtin_amdgcn_wmma_f32_16x16x4_f32(false, A, false, Bf,
                                                (short)0, c, false, false);
    }
    cur ^= 1;
  }

  const float bv = bias[n];
  float* orow = out + ((size_t)b * N + rowBase + (hi << 3)) * 16 + n;
#pragma unroll
  for (int i = 0; i < 8; ++i) orow[i * 16] = fmaxf(c[i] + bv, 0.0f);
}

// ---------------------------------------------------------------------------
extern "C" void kernel_launch(void* const* d_in, const int* in_sizes, int n_in,
                              void* d_out, int out_size, void* d_ws, size_t ws_size,
                              hipStream_t stream) {
  const float* x  = (const float*)d_in[0];   // [B,N,F]
  const float* a  = (const float*)d_in[1];   // [B,N,N]
  const float* W1 = (const float*)d_in[2];   // [F,H]
  const float* b1 = (const float*)d_in[3];   // [H]
  const float* W2 = (const float*)d_in[4];   // [H,H]
  const float* b2 = (const float*)d_in[5];   // [H]
  float* out = (float*)d_out;

  const int B = 8, N = 4096, F = 128, H = 16;
  const int R = B * N;                       // 32768 rows total

  // scratch: h1p, h1, h2p  (each R*H floats = 2 MB; 6 MB total)
  float* h1p = (float*)d_ws;
  float* h1  = h1p + (size_t)R * H;
  float* h2p = h1  + (size_t)R * H;

  const int tiles  = R >> 4;                         // 2048
  const int blocks = (tiles + WAVES_PER_BLOCK - 1) / WAVES_PER_BLOCK;
  dim3 blk(32 * WAVES_PER_BLOCK);

  // layer 1
  proj16<<<blocks, blk, 0, stream>>>(x, W1, h1p, R, F);
  agg16 <<<blocks, blk, 0, stream>>>(a, h1p, b1, h1, N, tiles);
  // layer 2
  proj16<<<blocks, blk, 0, stream>>>(h1, W2, h2p, R, H);
  agg16 <<<blocks, blk, 0, stream>>>(a, h2p, b2, out, N, tiles);
}